// CrossAttentionPooling_57105885167635
// MI455X (gfx1250) — compile-verified
//
#include <hip/hip_runtime.h>

typedef __attribute__((ext_vector_type(2))) float v2f;
typedef __attribute__((ext_vector_type(8))) float v8f;
typedef int v4i_vs __attribute__((vector_size(16)));   // matches builtin's "V4i"

#define B_    4
#define H_    16
#define LQ_   256
#define LK_   2048
#define D_    64
#define HID_  1024
#define QT_   16
#define KT_   32
#define NCHUNK (LK_/KT_)   /* 64 */
#define KSTR  68           /* sK row stride (floats): banks 4k+d, conflict-free */
#define VSTR  80           /* sV row stride (floats): halves differ by 32 banks */
#define PSTR  36           /* sP row stride: 36m mod 64 = 4*(9m mod 16) -> permutation */

/* ---- CDNA5 async global->LDS path (ASYNCcnt), with safe fallback ---- */
#if defined(__has_builtin)
# if __has_builtin(__builtin_amdgcn_global_load_async_to_lds_b128) && \
     __has_builtin(__builtin_amdgcn_s_wait_asynccnt)
#  define ASYNC_LDS 1
# endif
#endif

__device__ __forceinline__ void async_copy_b128(const float* g, float* l) {
#ifdef ASYNC_LDS
    __builtin_amdgcn_global_load_async_to_lds_b128(
        (__attribute__((address_space(1))) v4i_vs*)g,
        (__attribute__((address_space(3))) v4i_vs*)l,
        0, 0);
#else
    *(float4*)l = *(const float4*)g;
#endif
}

#ifdef ASYNC_LDS
# define WAIT_ASYNC(n) __builtin_amdgcn_s_wait_asynccnt(n)
#else
# define WAIT_ASYNC(n)
#endif

__global__ __launch_bounds__(256)
void dba_l1_attn_kernel(const float* __restrict__ Qg,
                        const float* __restrict__ Kg,
                        const float* __restrict__ Vg,
                        const float* __restrict__ diagg,
                        float* __restrict__ outg,
                        float* __restrict__ attng)
{
    __shared__ __align__(16) float sK[2][KT_ * KSTR];  // 2 x  8,704 B
    __shared__ __align__(16) float sV[2][KT_ * VSTR];  // 2 x 10,240 B
    __shared__ __align__(16) float sP[QT_ * PSTR];     //  2,304 B
    __shared__ __align__(16) float sRed[4 * 256];      //  4,096 B
    __shared__ float sRow[QT_];

    const int t = threadIdx.x;
    const int w = t >> 5;                // wave 0..7
    const int l = t & 31;                // lane

    const int blk = blockIdx.x;          // b*256 + h*16 + qt
    const int qt  = blk & 15;
    const int h   = (blk >> 4) & 15;
    const int b   = blk >> 8;
    const int q0  = qt * QT_;

    const float CENTER = 64.0f * 1.12838f;                 // 72.21632
    const float invS   = 1.0f / (sqrtf(0.72676f) * 8.0f);

    const int qrow = t >> 4;             // p-stage q ownership
    const int ksub = t & 15;

    // Q row in VGPRs; diag is lane-uniform (compiler keeps it in SGPRs)
    float4 qv[16], dg[16];
    {
        const float* qp = Qg + ((size_t)(b * LQ_ + q0 + qrow)) * HID_ + h * D_;
        const float* dp = diagg + h * D_;
        #pragma unroll
        for (int d4 = 0; d4 < 16; ++d4) {
            qv[d4] = ((const float4*)qp)[d4];
            dg[d4] = ((const float4*)dp)[d4];
        }
    }

    if (t < QT_) sRow[t] = 0.0f;

    v8f  cacc = {};
    float psum = 0.0f;

    float* attnBase = attng + ((size_t)((b * H_ + h) * LQ_ + q0)) * LK_;

    const int myN  = (w & 3) * 16;          // n-tile of head dim
    const int myK0 = (w >> 2) * (KT_ / 2);  // k-half within chunk
    const int mlan = l & 15;
    const int kbof = 2 * (l >> 4);          // upper half-wave: K+2,K+3

    // per-chunk async issue: 2 K + 2 V b128 per thread = 4 ASYNCcnt ops
    auto issue_chunk = [&](int ch, int buf) {
        const size_t rowBase = ((size_t)(b * LK_ + ch * KT_)) * HID_ + h * D_;
        #pragma unroll
        for (int j = 0; j < 2; ++j) {
            int idx = j * 256 + t;          // float4 index within 32x16 chunk
            int r   = idx >> 4;
            int c4  = idx & 15;
            const float* gk = Kg + rowBase + (size_t)r * HID_ + c4 * 4;
            const float* gv = Vg + rowBase + (size_t)r * HID_ + c4 * 4;
            async_copy_b128(gk, &sK[buf][r * KSTR + c4 * 4]);
            async_copy_b128(gv, &sV[buf][r * VSTR + c4 * 4]);
        }
    };

    issue_chunk(0, 0);   // prime the pipeline

    for (int ch = 0; ch < NCHUNK; ++ch) {
        const int cur = ch & 1;

        // prefetch next chunk into the other buffer, then wait for current
        if (ch + 1 < NCHUNK) {
            issue_chunk(ch + 1, cur ^ 1);
            WAIT_ASYNC(4);        // newest 4 (prefetch) may stay in flight
        } else {
            WAIT_ASYNC(0);
        }
        __syncthreads();          // chunk `ch` visible to all waves

        // ---- p stage: L1 distance -> exp ----
        // 4 independent FMA chains (x/y/z/w): v_sub + v_fma(|abs|) = 2 VALU/elem
        #pragma unroll
        for (int i = 0; i < KT_ / 16; ++i) {   // 2 pairs per thread per chunk
            int k = ksub + 16 * i;
            float sx = 0.0f, sy = 0.0f, sz = 0.0f, sw = 0.0f;
            #pragma unroll
            for (int d4 = 0; d4 < 16; ++d4) {
                float4 kk4 = *(const float4*)&sK[cur][k * KSTR + d4 * 4];
                sx = fmaf(dg[d4].x, fabsf(kk4.x - qv[d4].x), sx);
                sy = fmaf(dg[d4].y, fabsf(kk4.y - qv[d4].y), sy);
                sz = fmaf(dg[d4].z, fabsf(kk4.z - qv[d4].z), sz);
                sw = fmaf(dg[d4].w, fabsf(kk4.w - qv[d4].w), sw);
            }
            float s = (sx + sy) + (sz + sw);
            float p = __expf((CENTER - s) * invS);
            sP[qrow * PSTR + k] = p;
            attnBase[(size_t)qrow * LK_ + ch * KT_ + k] = p;  // unnormalized
            psum += p;
        }
        __syncthreads();

        // ---- PV: V_WMMA_F32_16X16X4_F32, persistent accumulator ----
        #pragma unroll
        for (int kk = 0; kk < KT_ / 2; kk += 4) {   // 4 wmma per chunk per wave
            int kb = myK0 + kk + kbof;
            v2f a, bf;
            a.x  = sP[mlan * PSTR + kb];
            a.y  = sP[mlan * PSTR + kb + 1];
            bf.x = sV[cur][kb * VSTR + myN + mlan];
            bf.y = sV[cur][(kb + 1) * VSTR + myN + mlan];
            cacc = __builtin_amdgcn_wmma_f32_16x16x4_f32(
                       false, a, false, bf, (short)0, cacc, false, false);
        }
        __syncthreads();   // protect buf[cur] from next iteration's prefetch
    }

    atomicAdd(&sRow[qrow], psum);    // ds_add_f32
    __threadfence();                 // publish unnormalized p
    __syncthreads();

    // ---- reduce wave pairs (w, w+4 share n-tile, differ in k-half) ----
    if (w >= 4) {
        #pragma unroll
        for (int r = 0; r < 8; ++r) sRed[(w - 4) * 256 + r * 32 + l] = cacc[r];
    }
    __syncthreads();
    if (w < 4) {
        #pragma unroll
        for (int r = 0; r < 8; ++r) {
            float v = cacc[r] + sRed[w * 256 + r * 32 + l];
            int m = r + 8 * (l >> 4);       // C/D layout: lanes 16-31 hold M=r+8
            int n = l & 15;
            float inv = 1.0f / sRow[m];
            outg[((size_t)(b * LQ_ + q0 + m)) * HID_ + h * D_ + myN + n] = v * inv;
        }
    }

    // ---- normalize this WG's attention slice in place (L2-resident) ----
    __syncthreads();
    for (int qr = 0; qr < QT_; ++qr) {
        float inv = 1.0f / sRow[qr];
        #pragma unroll
        for (int i = 0; i < LK_ / 256; ++i) {
            size_t pos = (size_t)qr * LK_ + i * 256 + t;
            attnBase[pos] *= inv;
        }
    }
}

extern "C" void kernel_launch(void* const* d_in, const int* in_sizes, int n_in,
                              void* d_out, int out_size, void* d_ws, size_t ws_size,
                              hipStream_t stream) {
    const float* Q    = (const float*)d_in[0];   // [4,256,1024]
    const float* K    = (const float*)d_in[1];   // [4,2048,1024]
    const float* V    = (const float*)d_in[2];   // [4,2048,1024]
    const float* diag = (const float*)d_in[3];   // [16,1,64]

    float* out  = (float*)d_out;                          // [4,256,1024]
    float* attn = out + (size_t)B_ * LQ_ * HID_;          // [4,16,256,2048]

    dim3 grid(B_ * H_ * (LQ_ / QT_));   // 1024 workgroups
    dim3 block(256);                    // 8 wave32
    dba_l1_attn_kernel<<<grid, block, 0, stream>>>(Q, K, V, diag, out, attn);
}